// GraphEncoder_61272003445438
// MI455X (gfx1250) — compile-verified
//
#include <hip/hip_runtime.h>
#include <hip/hip_bf16.h>

typedef float v2f __attribute__((ext_vector_type(2)));
typedef float v8f __attribute__((ext_vector_type(8)));

#define NND 50000      // nodes
#define NRL 8          // relations
#define DIM 128        // in = hid = out dim
#define NEG 800000     // edges

// ---------------- zero fill (float4 stores) ----------------
__global__ void zero_f4_kernel(float4* __restrict__ p, int n4) {
    int i = blockIdx.x * blockDim.x + threadIdx.x;
    if (i < n4) p[i] = make_float4(0.f, 0.f, 0.f, 0.f);
}

// ---------------- edge scatter: one wave32 per edge ----------------
// sums[r*N + t][0:128] += x[h][0:128]  (f32 hw atomics), cnt[r*N+t] += 1
__global__ void scatter_kernel(const float* __restrict__ x,
                               const int* __restrict__ h,
                               const int* __restrict__ r,
                               const int* __restrict__ t,
                               float* __restrict__ sums,
                               float* __restrict__ cnt,
                               int nedges) {
    const int e    = blockIdx.x * 8 + (threadIdx.x >> 5);
    const int lane = threadIdx.x & 31;
    if (e >= nedges) return;
    const int hh = h[e];
    const int rr = r[e];
    const int tt = t[e];
    const float4 v = ((const float4*)(x + (size_t)hh * DIM))[lane];
    float* dst = sums + ((size_t)rr * NND + tt) * DIM + lane * 4;
    unsafeAtomicAdd(dst + 0, v.x);
    unsafeAtomicAdd(dst + 1, v.y);
    unsafeAtomicAdd(dst + 2, v.z);
    unsafeAtomicAdd(dst + 3, v.w);
    if (lane == 0) unsafeAtomicAdd(cnt + (size_t)rr * NND + tt, 1.0f);
}

// ---------------- seed d_out with bias (layer-2 mean gets += on top) ----
__global__ void init_out_kernel(float* __restrict__ dout,
                                const float* __restrict__ b) {
    int j = threadIdx.x;
    if (j < DIM) dout[j] = b[j];
}

// ---------------- fused mean + RGCN GEMM via V_WMMA_F32_16X16X4_F32 -----
// out[n,:] = sum_r (sums[r,n,:]/max(cnt,1)) @ W[r] + xin[n,:] @ root (+bias)
// block: 256 threads = 8 waves; block handles 16 rows, wave w handles cols
// [16w, 16w+16).  mode 0: xout = relu(out + bias).  mode 1: atomic-add
// column sums / NND into dout (bias pre-seeded).
__global__ void rgcn_gemm_kernel(const float* __restrict__ sums,
                                 const float* __restrict__ cnt,
                                 const float* __restrict__ xin,
                                 const float* __restrict__ W,     // [8,128,128]
                                 const float* __restrict__ root,  // [128,128]
                                 const float* __restrict__ bias,  // [128]
                                 float* __restrict__ xout,
                                 float* __restrict__ dout,
                                 int mode) {
    const int n0   = blockIdx.x * 16;
    const int wid  = threadIdx.x >> 5;
    const int lane = threadIdx.x & 31;
    const int half = lane >> 4;       // 0: K pair {0,1}, 1: K pair {2,3}
    const int l15  = lane & 15;
    const int row  = n0 + l15;        // A-matrix row for this lane
    const int col  = wid * 16 + l15;  // B/C/D column for this lane
    const int kh   = half * 2;

    v8f c0 = {};
    v8f c1 = {};

    // 8 relation segments: A = sums[r]/max(cnt,1), B = W[r]
    for (int rr = 0; rr < NRL; ++rr) {
        const float* A = sums + ((size_t)rr * NND + row) * DIM;
        const float  invc = 1.0f / fmaxf(cnt[(size_t)rr * NND + row], 1.0f);
        const float* B = W + (size_t)rr * DIM * DIM;
#pragma unroll 4
        for (int k0 = 0; k0 < DIM; k0 += 8) {
            const float2 av0 = *(const float2*)(A + k0 + kh);
            const float2 av1 = *(const float2*)(A + k0 + 4 + kh);
            v2f a0; a0.x = av0.x * invc; a0.y = av0.y * invc;
            v2f a1; a1.x = av1.x * invc; a1.y = av1.y * invc;
            v2f b0; b0.x = B[(k0 + kh    ) * DIM + col];
                    b0.y = B[(k0 + kh + 1) * DIM + col];
            v2f b1; b1.x = B[(k0 + kh + 4) * DIM + col];
                    b1.y = B[(k0 + kh + 5) * DIM + col];
            c0 = __builtin_amdgcn_wmma_f32_16x16x4_f32(false, a0, false, b0,
                                                       (short)0, c0, false, false);
            c1 = __builtin_amdgcn_wmma_f32_16x16x4_f32(false, a1, false, b1,
                                                       (short)0, c1, false, false);
        }
    }
    // root segment: A = xin, B = root
    {
        const float* A = xin + (size_t)row * DIM;
#pragma unroll 4
        for (int k0 = 0; k0 < DIM; k0 += 8) {
            const float2 av0 = *(const float2*)(A + k0 + kh);
            const float2 av1 = *(const float2*)(A + k0 + 4 + kh);
            v2f a0; a0.x = av0.x; a0.y = av0.y;
            v2f a1; a1.x = av1.x; a1.y = av1.y;
            v2f b0; b0.x = root[(k0 + kh    ) * DIM + col];
                    b0.y = root[(k0 + kh + 1) * DIM + col];
            v2f b1; b1.x = root[(k0 + kh + 4) * DIM + col];
                    b1.y = root[(k0 + kh + 5) * DIM + col];
            c0 = __builtin_amdgcn_wmma_f32_16x16x4_f32(false, a0, false, b0,
                                                       (short)0, c0, false, false);
            c1 = __builtin_amdgcn_wmma_f32_16x16x4_f32(false, a1, false, b1,
                                                       (short)0, c1, false, false);
        }
    }

    if (mode == 0) {
        // xout[n0+m, col] = relu(c[m] + bias[col]);  m = v + half*8
        const float bj = bias[col];
#pragma unroll
        for (int v = 0; v < 8; ++v) {
            const int m = v + half * 8;
            float val = c0[v] + c1[v] + bj;
            val = fmaxf(val, 0.0f);
            xout[(size_t)(n0 + m) * DIM + col] = val;
        }
    } else {
        // column partial sum over this tile's 16 rows -> dout (mean)
        float s = 0.0f;
#pragma unroll
        for (int v = 0; v < 8; ++v) s += c0[v] + c1[v];
        s += __shfl_down(s, 16, 32);   // fold rows 8..15 half into rows 0..7 half
        if (half == 0)
            unsafeAtomicAdd(&dout[col], s * (1.0f / (float)NND));
    }
}

extern "C" void kernel_launch(void* const* d_in, const int* in_sizes, int n_in,
                              void* d_out, int out_size, void* d_ws, size_t ws_size,
                              hipStream_t stream) {
    const int*   h     = (const int*)d_in[0];
    const int*   r     = (const int*)d_in[1];
    const int*   t     = (const int*)d_in[2];
    const float* x_emb = (const float*)d_in[3];
    const float* W1    = (const float*)d_in[4];
    const float* root1 = (const float*)d_in[5];
    const float* b1    = (const float*)d_in[6];
    const float* W2    = (const float*)d_in[7];
    const float* root2 = (const float*)d_in[8];
    const float* b2    = (const float*)d_in[9];
    float*       dout  = (float*)d_out;

    // workspace layout (f32): sums[8*N*128] | cnt[8*N] | x1[N*128]
    float* sums = (float*)d_ws;
    float* cnt  = sums + (size_t)NRL * NND * DIM;
    float* x1   = cnt  + (size_t)NRL * NND;

    const size_t zeroN = (size_t)NRL * NND * DIM + (size_t)NRL * NND; // 51.6M
    const int n4 = (int)(zeroN / 4);
    const int zb = (n4 + 255) / 256;

    const int E  = in_sizes[0];
    const int sb = (E + 7) / 8;
    const int gb = NND / 16;   // 3125 exactly

    // ---- layer 1 ----
    zero_f4_kernel<<<zb, 256, 0, stream>>>((float4*)sums, n4);
    scatter_kernel<<<sb, 256, 0, stream>>>(x_emb, h, r, t, sums, cnt, E);
    rgcn_gemm_kernel<<<gb, 256, 0, stream>>>(sums, cnt, x_emb, W1, root1, b1,
                                             x1, nullptr, 0);
    // ---- layer 2 ----
    zero_f4_kernel<<<zb, 256, 0, stream>>>((float4*)sums, n4);
    scatter_kernel<<<sb, 256, 0, stream>>>(x1, h, r, t, sums, cnt, E);
    init_out_kernel<<<1, 128, 0, stream>>>(dout, b2);
    rgcn_gemm_kernel<<<gb, 256, 0, stream>>>(sums, cnt, x1, W2, root2, b1,
                                             nullptr, dout, 1);
}